// BaselineModelMoreLogic_69861938036761
// MI455X (gfx1250) — compile-verified
//
#include <hip/hip_runtime.h>
#include <hip/hip_bf16.h>
#include <math.h>

typedef float v2f __attribute__((ext_vector_type(2)));
typedef float v4f __attribute__((ext_vector_type(4)));
typedef float v8f __attribute__((ext_vector_type(8)));
typedef int   v4i __attribute__((ext_vector_type(4)));

#define MTOT   64      // batch rows (all fit in one block tile)
#define NOUT   4096    // output dim of every GEMM
#define NTILE  128     // N columns per block (8 waves x 16)
#define KB     32      // K chunk staged in LDS
#define LDSS   36      // padded LDS row stride in floats: 144B rows (16B aligned),
                       // 36*m mod 64 hits all multiples of 4 -> conflict-free frag reads

// ---- gfx1250 async global->LDS path (guarded: falls back to staged copy) ----
#if __has_builtin(__builtin_amdgcn_global_load_async_to_lds_b128)
#define USE_ASYNC_LDS 1
typedef v4i __attribute__((address_space(1)))* as1_v4i;   // global int4*
typedef v4i __attribute__((address_space(3)))* as3_v4i;   // LDS int4*
#else
#define USE_ASYNC_LDS 0
#endif

__device__ __forceinline__ void copy16_to_lds(const float* __restrict__ g, float* l)
{
#if USE_ASYNC_LDS
    __builtin_amdgcn_global_load_async_to_lds_b128((as1_v4i)g, (as3_v4i)l, 0, 0);
#else
    *(v4f*)l = *(const v4f*)g;
#endif
}

__device__ __forceinline__ void wait_async_lds()
{
#if USE_ASYNC_LDS
#if __has_builtin(__builtin_amdgcn_s_wait_asynccnt)
    __builtin_amdgcn_s_wait_asynccnt(0);
#else
    asm volatile("s_wait_asynccnt 0x0" ::: "memory");
#endif
#endif
}

struct GemmArgs {
    const float* X[3];     // activations [64, K]
    const float* W[3];     // weights [4096, K] row-major (we need X @ W^T)
    const float* bias[3];  // [4096]
    const float* E[3];     // elementwise gate source h (stage 2 only)
    float*       O[3];     // output [64, 4096]
    float*       SS;       // per-row sum-of-squares base (stage 2 only), 3*64
    int          K[3];
    int          gate;     // 0: store h ; 1: y = E * sigmoid(acc), accumulate SS
};

// ---------------------------------------------------------------------------
// Prep: zero SS, text max-pool over L=30, audio ragged mean over valid frames
// ---------------------------------------------------------------------------
__global__ __launch_bounds__(256) void prep_pool(
    const float* __restrict__ text,     // [64,30,4096]
    const float* __restrict__ audio,    // [64,1024,128]
    const int*   __restrict__ nframes,  // [64]
    const int*   __restrict__ rawlen,   // [1]
    float* __restrict__ xt,             // [64,4096]
    float* __restrict__ xa,             // [64,1024]
    float* __restrict__ SS)             // [192]
{
    const int idx = blockIdx.x * 256 + threadIdx.x;

    if (idx < 3 * MTOT) SS[idx] = 0.0f;

    if (idx < MTOT * 4096) {                        // text max-pool
        const int b = idx >> 12, d = idx & 4095;
        const float* p = text + (size_t)b * 30 * 4096 + d;
        float m = p[0];
        #pragma unroll 4
        for (int l = 1; l < 30; ++l) m = fmaxf(m, p[(size_t)l * 4096]);
        xt[idx] = m;
    }

    if (idx < MTOT * 1024) {                        // audio masked mean
        const int b = idx >> 10, c = idx & 1023;
        const int ratio = (int)roundf((float)rawlen[0] / 128.0f);   // = 16
        int nf = (int)((float)nframes[b] / (float)ratio);           // trunc, matches ref
        if (nf < 1) nf = 1;
        if (nf > 128) nf = 128;
        const float* p = audio + ((size_t)b * 1024 + c) * 128;
        float s = 0.0f;
        for (int t = 0; t < nf; ++t) s += p[t];
        xa[idx] = s / (float)nf;
    }
}

// ---------------------------------------------------------------------------
// Stage one K-chunk (X: 64xKB, W: 128xKB) into one LDS buffer.
// 6 x b128 per thread; async DMA to LDS on gfx1250, staged copy otherwise.
// ---------------------------------------------------------------------------
__device__ __forceinline__ void stage_chunk(const float* __restrict__ X,
                                            const float* __restrict__ W,
                                            int K, int n0, int k0,
                                            float* xbuf, float* wbuf, int t)
{
    #pragma unroll
    for (int it = 0; it < 2; ++it) {            // X tile: 512 float4
        int f = t + it * 256;
        int row = f >> 3, cv = (f & 7) << 2;
        copy16_to_lds(X + (size_t)row * K + k0 + cv, xbuf + row * LDSS + cv);
    }
    #pragma unroll
    for (int it = 0; it < 4; ++it) {            // W tile: 1024 float4
        int f = t + it * 256;
        int row = f >> 3, cv = (f & 7) << 2;
        copy16_to_lds(W + (size_t)(n0 + row) * K + k0 + cv, wbuf + row * LDSS + cv);
    }
}

// ---------------------------------------------------------------------------
// Fused 3-branch GEMM: H = X @ W^T + bias   (fp32 WMMA 16x16x4)
// grid = (NOUT/NTILE, 3), block = 256 (8 wave32)
// Each wave owns one 16-wide N strip; 4 M-subtiles cover all 64 rows.
// Double-buffered LDS: chunk ko+1 streams in (async DMA) while ko computes.
// ---------------------------------------------------------------------------
__global__ __launch_bounds__(256) void geu_gemm(GemmArgs a)
{
    const int br = blockIdx.y;
    const float* __restrict__ X    = a.X[br];
    const float* __restrict__ W    = a.W[br];
    const float* __restrict__ bias = a.bias[br];
    const int K  = a.K[br];
    const int n0 = blockIdx.x * NTILE;

    __shared__ float Xs[2][MTOT  * LDSS];   // 2 x  9216 B
    __shared__ float Ws[2][NTILE * LDSS];   // 2 x 18432 B  (total 55296 B)

    const int t    = threadIdx.x;
    const int wid  = t >> 5;
    const int lane = t & 31;
    const int hf   = lane >> 4;   // which 16-lane half
    const int r    = lane & 15;
    const int ncol = n0 + wid * 16 + r;

    // C/D layout: vgpr i, half hf -> row M = i + 8*hf, col N = r.
    // All 8 accum elements of a lane share column ncol -> splat bias.
    const float bval = bias[ncol];
    v8f acc[4];
    #pragma unroll
    for (int mt = 0; mt < 4; ++mt)
        acc[mt] = v8f{bval, bval, bval, bval, bval, bval, bval, bval};

    const int nk = K / KB;
    stage_chunk(X, W, K, n0, 0, Xs[0], Ws[0], t);

    for (int ko = 0; ko < nk; ++ko) {
        const int cur = ko & 1;
        wait_async_lds();          // my chunk-ko DMA complete (async path)
        __syncthreads();           // everyone's chunk ko visible, buf cur^1 free
        if (ko + 1 < nk)
            stage_chunk(X, W, K, n0, (ko + 1) * KB, Xs[cur ^ 1], Ws[cur ^ 1], t);

        const float* xs = Xs[cur];
        const float* ws = Ws[cur];
        // A frag: lane -> M = r (within mt tile), K = 2*hf + vgpr
        // B frag: lane -> N = r (within wave strip), K = 2*hf + vgpr
        #pragma unroll
        for (int kk = 0; kk < KB; kk += 4) {
            v2f bfrag = *(const v2f*)(ws + (wid * 16 + r) * LDSS + kk + 2 * hf);
            #pragma unroll
            for (int mt = 0; mt < 4; ++mt) {
                v2f afrag = *(const v2f*)(xs + (mt * 16 + r) * LDSS + kk + 2 * hf);
                acc[mt] = __builtin_amdgcn_wmma_f32_16x16x4_f32(
                    false, afrag, false, bfrag, (short)0, acc[mt], false, false);
            }
        }
    }

    float* __restrict__ O = a.O[br];
    if (!a.gate) {
        #pragma unroll
        for (int mt = 0; mt < 4; ++mt)
            #pragma unroll
            for (int i = 0; i < 8; ++i) {
                int row = mt * 16 + i + 8 * hf;
                O[(size_t)row * NOUT + ncol] = acc[mt][i];
            }
    } else {
        const float* __restrict__ E = a.E[br];
        float* __restrict__ SS = a.SS + br * MTOT;
        #pragma unroll
        for (int mt = 0; mt < 4; ++mt) {
            #pragma unroll
            for (int i = 0; i < 8; ++i) {
                int row = mt * 16 + i + 8 * hf;
                float h = E[(size_t)row * NOUT + ncol];
                float g = acc[mt][i];
                float y = h * (1.0f / (1.0f + __expf(-g)));
                O[(size_t)row * NOUT + ncol] = y;
                // reduce y^2 across the 16 column-lanes of this half-wave
                float v = y * y;
                #pragma unroll
                for (int m = 8; m >= 1; m >>= 1) v += __shfl_xor(v, m, 16);
                if (r == 0) atomicAdd(&SS[row], v);
            }
        }
    }
}

// ---------------------------------------------------------------------------
// Normalize: out = y / max(||y||, eps), ss index = idx / 4096
// ---------------------------------------------------------------------------
__global__ __launch_bounds__(256) void finalize_norm(float* __restrict__ out,
                                                     const float* __restrict__ SS)
{
    const int idx = blockIdx.x * 256 + threadIdx.x;
    if (idx < 3 * MTOT * NOUT) {
        const int rowid = idx >> 12;                // branch*64 + b
        const float d = fmaxf(sqrtf(SS[rowid]), 1e-12f);
        out[idx] = out[idx] / d;
    }
}

// ---------------------------------------------------------------------------
extern "C" void kernel_launch(void* const* d_in, const int* in_sizes, int n_in,
                              void* d_out, int out_size, void* d_ws, size_t ws_size,
                              hipStream_t stream)
{
    const float* text    = (const float*)d_in[0];
    const float* video   = (const float*)d_in[1];
    const float* audio   = (const float*)d_in[2];
    const int*   nframes = (const int*)d_in[3];
    const int*   rawlen  = (const int*)d_in[4];
    const float* Wt  = (const float*)d_in[5],  *bt  = (const float*)d_in[6];
    const float* Wgt = (const float*)d_in[7],  *bgt = (const float*)d_in[8];
    const float* Wv  = (const float*)d_in[9],  *bv  = (const float*)d_in[10];
    const float* Wgv = (const float*)d_in[11], *bgv = (const float*)d_in[12];
    const float* Wa  = (const float*)d_in[13], *ba  = (const float*)d_in[14];
    const float* Wga = (const float*)d_in[15], *bga = (const float*)d_in[16];

    float* ws = (float*)d_ws;
    float* xt = ws;                        // 64*4096
    float* xa = xt + MTOT * 4096;          // 64*1024
    float* ht = xa + MTOT * 1024;          // 64*4096
    float* hv = ht + MTOT * 4096;          // 64*4096
    float* ha = hv + MTOT * 4096;          // 64*4096
    float* SS = ha + MTOT * 4096;          // 192

    float* out = (float*)d_out;
    float* yt = out;
    float* yv = out + MTOT * NOUT;
    float* ya = out + 2 * MTOT * NOUT;

    prep_pool<<<1024, 256, 0, stream>>>(text, audio, nframes, rawlen, xt, xa, SS);

    GemmArgs g1;
    g1.X[0] = xt;  g1.X[1] = video; g1.X[2] = xa;
    g1.W[0] = Wt;  g1.W[1] = Wv;    g1.W[2] = Wa;
    g1.bias[0] = bt; g1.bias[1] = bv; g1.bias[2] = ba;
    g1.E[0] = g1.E[1] = g1.E[2] = nullptr;
    g1.O[0] = ht;  g1.O[1] = hv;    g1.O[2] = ha;
    g1.SS = nullptr;
    g1.K[0] = 4096; g1.K[1] = 4096; g1.K[2] = 1024;
    g1.gate = 0;
    geu_gemm<<<dim3(NOUT / NTILE, 3), 256, 0, stream>>>(g1);

    GemmArgs g2;
    g2.X[0] = ht;  g2.X[1] = hv;    g2.X[2] = ha;
    g2.W[0] = Wgt; g2.W[1] = Wgv;   g2.W[2] = Wga;
    g2.bias[0] = bgt; g2.bias[1] = bgv; g2.bias[2] = bga;
    g2.E[0] = ht;  g2.E[1] = hv;    g2.E[2] = ha;
    g2.O[0] = yt;  g2.O[1] = yv;    g2.O[2] = ya;
    g2.SS = SS;
    g2.K[0] = 4096; g2.K[1] = 4096; g2.K[2] = 4096;
    g2.gate = 1;
    geu_gemm<<<dim3(NOUT / NTILE, 3), 256, 0, stream>>>(g2);

    finalize_norm<<<(3 * MTOT * NOUT + 255) / 256, 256, 0, stream>>>(out, SS);
}